// SparseActivationLinear_13237089206317
// MI455X (gfx1250) — compile-verified
//
#include <hip/hip_runtime.h>

// ---------------------------------------------------------------------------
// SparseActivationLinear on MI455X (gfx1250):
//   out[M,N] = sparsify_2:4(x)[M,K] @ W[N,K]^T + bias,  M=8192, N=K=4096
// v3: SWMMAC bf16 core + async global->LDS staging (double-buffered LDS),
// W pre-converted to bf16 so all staging is pure byte-copy.
// ---------------------------------------------------------------------------

#define K_FEAT 4096
#define N_FEAT 4096
#define M_TOTAL 8192  // 4 * 2048

typedef __bf16 v16bf   __attribute__((ext_vector_type(16)));
typedef __bf16 v32bf   __attribute__((ext_vector_type(32)));
typedef __bf16 bf16x8  __attribute__((ext_vector_type(8)));
typedef __bf16 bf16x16 __attribute__((ext_vector_type(16)));
typedef float  v8f     __attribute__((ext_vector_type(8)));
typedef int    i32x4   __attribute__((vector_size(16)));  // matches builtin proto

#define AS1 __attribute__((address_space(1)))
#define AS3 __attribute__((address_space(3)))

#if defined(__has_builtin)
#if __has_builtin(__builtin_amdgcn_global_load_async_to_lds_b128) && \
    __has_builtin(__builtin_amdgcn_global_load_async_to_lds_b32)
#define HAVE_ASYNC 1
#endif
#endif

#ifdef HAVE_ASYNC
#if __has_builtin(__builtin_amdgcn_s_wait_asynccnt)
#define WAIT_ASYNCCNT(n) __builtin_amdgcn_s_wait_asynccnt(n)
#else
#define WAIT_ASYNCCNT(n) asm volatile("s_wait_asynccnt " #n ::: "memory")
#endif
// Builtin proto (from diagnostics): (AS1 v4i* gsrc, AS3 v4i* ldst, Ii offset, Ii cpol)
#define ASYNC_B128(g, l)                                  \
  __builtin_amdgcn_global_load_async_to_lds_b128(         \
      (AS1 i32x4*)(g), (AS3 i32x4*)(l), 0, 0)
#define ASYNC_B32(g, l)                                   \
  __builtin_amdgcn_global_load_async_to_lds_b32(          \
      (AS1 int*)(g), (AS3 int*)(l), 0, 0)
#else
#define WAIT_ASYNCCNT(n)
#endif

static __device__ __forceinline__ unsigned short f2bf(float f) {
  unsigned int u = __builtin_bit_cast(unsigned int, f);
  unsigned int r = u + 0x7FFFu + ((u >> 16) & 1u);  // round-to-nearest-even
  return (unsigned short)(r >> 16);
}

// ---------------------------------------------------------------------------
// Kernel 0: W fp32 -> bf16 (one-shot; removes VALU cvt from GEMM hot loop and
// halves W read traffic there; Wb (32 MB) stays L2-resident).
// ---------------------------------------------------------------------------
__global__ __launch_bounds__(256) void convert_w_kernel(
    const float* __restrict__ W, unsigned short* __restrict__ Wb) {
  size_t i = (size_t)(blockIdx.x * 256u + threadIdx.x) * 8u;
  float4 f0 = *(const float4*)(W + i);
  float4 f1 = *(const float4*)(W + i + 4);
  alignas(16) unsigned short h[8] = {f2bf(f0.x), f2bf(f0.y), f2bf(f0.z), f2bf(f0.w),
                                     f2bf(f1.x), f2bf(f1.y), f2bf(f1.z), f2bf(f1.w)};
  *(uint4*)(Wb + i) = *(const uint4*)h;
}

// ---------------------------------------------------------------------------
// Kernel 1: 2:4 sparsify + compress to bf16 + pack 2-bit index metadata
// (one uint32 per 32 expanded K values, exactly the SWMMAC index-VGPR layout).
// ---------------------------------------------------------------------------
__global__ __launch_bounds__(256) void sparsify_compress_kernel(
    const float* __restrict__ x, unsigned short* __restrict__ Ap,
    unsigned int* __restrict__ meta) {
  const unsigned gid = blockIdx.x * 256u + threadIdx.x;  // 0 .. M*128-1
  const unsigned row = gid >> 7;
  const unsigned seg = gid & 127u;
  const float* src = x + (size_t)row * K_FEAT + seg * 32u;

  alignas(16) unsigned short packed[16];
  unsigned int code = 0;
#pragma unroll
  for (int g = 0; g < 8; ++g) {
    float4 v = *(const float4*)(src + g * 4);
    float a[4]  = {v.x, v.y, v.z, v.w};
    float ab[4] = {fabsf(v.x), fabsf(v.y), fabsf(v.z), fabsf(v.w)};
    int i0 = 0;  // argmax |.| (ties -> lower index, matches jax top_k)
#pragma unroll
    for (int i = 1; i < 4; ++i)
      if (ab[i] > ab[i0]) i0 = i;
    int i1 = (i0 == 0) ? 1 : 0;
#pragma unroll
    for (int i = 0; i < 4; ++i)
      if (i != i0 && ab[i] > ab[i1]) i1 = i;
    int lo = i0 < i1 ? i0 : i1;
    int hi = i0 < i1 ? i1 : i0;
    packed[2 * g]     = f2bf(a[lo]);
    packed[2 * g + 1] = f2bf(a[hi]);
    code |= ((unsigned)lo | ((unsigned)hi << 2)) << (4 * g);
  }
  unsigned short* dst = Ap + (size_t)row * (K_FEAT / 2) + seg * 16u;
  *(uint4*)(dst)     = *(const uint4*)&packed[0];
  *(uint4*)(dst + 8) = *(const uint4*)&packed[8];
  meta[(size_t)row * (K_FEAT / 32) + seg] = code;
}

// ---------------------------------------------------------------------------
// Kernel 2: SWMMAC bf16 GEMM, async double-buffered staging.
// 128x128 C-tile per 256-thread WG (8 wave32 waves in 2(M) x 4(N)),
// each wave 64x32 = 4x2 tiles, K-step 64 expanded / 32 compressed.
// Per wave per tile: 7 async ops (A:2 b128, meta:1 b32, B:4 b128).
// ---------------------------------------------------------------------------
#define BM 128
#define BN 128
#define BK 64   // expanded K per step
#define BKC 32  // compressed K per step
#define NSTEP (K_FEAT / BK)

__global__ __launch_bounds__(256) void swmmac_gemm_kernel(
    const unsigned short* __restrict__ Ap, const unsigned int* __restrict__ meta,
    const unsigned short* __restrict__ Wb, const float* __restrict__ bias,
    float* __restrict__ out) {
  __shared__ unsigned short sA[2][BM * BKC];   // 2 x 8 KB  [m][kc]
  __shared__ unsigned int   sMeta[2][BM * 2];  // 2 x 1 KB
  __shared__ unsigned short sB[2][BN * BK];    // 2 x 16 KB [n][k]

  const int tid  = threadIdx.x;
  const int m0   = blockIdx.y * BM;
  const int n0   = blockIdx.x * BN;
  const int lane = tid & 31;
  const int wave = tid >> 5;
  const int wm   = wave & 1;   // M sub-tile (64 rows)
  const int wn   = wave >> 1;  // N sub-tile (32 cols)
  const int lrow = lane & 15;
  const int lhi  = lane >> 4;  // K-half selector per ISA layouts

  // ---- precomputed per-thread staging pointers (advance by constants) ----
  // A chunks i = tid, tid+256 : row=i>>2, c=i&3 (16B each)
  const int ar0 = tid >> 2, ac0 = tid & 3;
  const unsigned short* gA0 = Ap + (size_t)(m0 + ar0) * (K_FEAT / 2) + ac0 * 8;
  const unsigned short* gA1 = gA0 + (size_t)64 * (K_FEAT / 2);  // row+64
  const int aoff0 = ar0 * BKC + ac0 * 8;
  const int aoff1 = aoff0 + 64 * BKC;
  // meta: row=tid>>1, wsel=tid&1 (4B each)
  const int mr = tid >> 1, mw = tid & 1;
  const unsigned int* gM = meta + (size_t)(m0 + mr) * (K_FEAT / 32) + mw;
  const int moff = mr * 2 + mw;
  // B chunks i = tid + j*256 (j=0..3): n=i>>3, kq=i&7 (16B each)
  const unsigned short* gB[4];
  int boff[4];
#pragma unroll
  for (int j = 0; j < 4; ++j) {
    int i = tid + j * 256;
    int n = i >> 3, kq = i & 7;
    gB[j] = Wb + (size_t)(n0 + n) * K_FEAT + kq * 8;
    boff[j] = n * BK + kq * 8;
  }

  auto stage = [&](int buf, int ks) {
    const size_t kcc = (size_t)ks * BKC;  // compressed-K advance
    const size_t kce = (size_t)ks * BK;   // expanded-K advance
#ifdef HAVE_ASYNC
    ASYNC_B128(gA0 + kcc, &sA[buf][aoff0]);
    ASYNC_B128(gA1 + kcc, &sA[buf][aoff1]);
    ASYNC_B32(gM + (size_t)ks * 2, &sMeta[buf][moff]);
#pragma unroll
    for (int j = 0; j < 4; ++j) ASYNC_B128(gB[j] + kce, &sB[buf][boff[j]]);
#else
    *(uint4*)&sA[buf][aoff0] = *(const uint4*)(gA0 + kcc);
    *(uint4*)&sA[buf][aoff1] = *(const uint4*)(gA1 + kcc);
    sMeta[buf][moff] = gM[(size_t)ks * 2];
#pragma unroll
    for (int j = 0; j < 4; ++j)
      *(uint4*)&sB[buf][boff[j]] = *(const uint4*)(gB[j] + kce);
#endif
  };

  v8f acc[4][2] = {};

  stage(0, 0);
  for (int ks = 0; ks < NSTEP; ++ks) {
    const int buf = ks & 1;
    if (ks + 1 < NSTEP) {
      stage(buf ^ 1, ks + 1);
      WAIT_ASYNCCNT(7);  // in-order: oldest 7 (current tile) have landed
    } else {
      WAIT_ASYNCCNT(0);
    }
    __syncthreads();

    // ---- B fragments: 64x16 bf16 per ISA layout ----
    union BF { v32bf v; bf16x16 h[2]; } b[2];
#pragma unroll
    for (int nn = 0; nn < 2; ++nn) {
      int col = wn * 32 + nn * 16 + lrow;
      const unsigned short* p = &sB[buf][col * BK + lhi * 16];
      b[nn].h[0] = *(const bf16x16*)p;         // K = lhi*16 .. +15
      b[nn].h[1] = *(const bf16x16*)(p + 32);  // K = lhi*16+32 .. +47
    }
    // ---- A fragments (compressed 16x32) + index word, then SWMMAC ----
#pragma unroll
    for (int mt = 0; mt < 4; ++mt) {
      int row = wm * 64 + mt * 16 + lrow;
      union AF { v16bf v; bf16x8 h[2]; } a;
      const unsigned short* p = &sA[buf][row * BKC + lhi * 8];
      a.h[0] = *(const bf16x8*)p;
      a.h[1] = *(const bf16x8*)(p + 16);
      int iv = (int)sMeta[buf][row * 2 + lhi];
#pragma unroll
      for (int nn = 0; nn < 2; ++nn) {
        acc[mt][nn] = __builtin_amdgcn_swmmac_f32_16x16x64_bf16(
            false, a.v, false, b[nn].v, acc[mt][nn], iv, false, false);
      }
    }
    __syncthreads();  // all waves done reading buf before it is re-staged
  }

  // ---- epilogue: C layout — VGPR r: lanes0-15 M=r, lanes16-31 M=r+8; N=lane%16
#pragma unroll
  for (int nn = 0; nn < 2; ++nn) {
    int n = n0 + wn * 32 + nn * 16 + lrow;
    float bv = bias[n];
#pragma unroll
    for (int mt = 0; mt < 4; ++mt) {
      int mbase = m0 + wm * 64 + mt * 16 + lhi * 8;
#pragma unroll
      for (int r = 0; r < 8; ++r) {
        out[(size_t)(mbase + r) * N_FEAT + n] = acc[mt][nn][r] + bv;
      }
    }
  }
}

// ---------------------------------------------------------------------------
extern "C" void kernel_launch(void* const* d_in, const int* in_sizes, int n_in,
                              void* d_out, int out_size, void* d_ws, size_t ws_size,
                              hipStream_t stream) {
  const float* x    = (const float*)d_in[0];  // (4,2048,4096) fp32
  const float* w    = (const float*)d_in[1];  // (4096,4096) fp32
  const float* bias = (const float*)d_in[2];  // (4096,) fp32
  float* out = (float*)d_out;                 // (4,2048,4096) fp32

  char* ws = (char*)d_ws;
  unsigned short* Ap = (unsigned short*)ws;                          // 32 MB
  unsigned int* meta = (unsigned int*)(ws + ((size_t)32 << 20));     // 4 MB
  unsigned short* Wb = (unsigned short*)(ws + ((size_t)36 << 20));   // 32 MB

  convert_w_kernel<<<dim3((N_FEAT * (size_t)K_FEAT / 8) / 256), 256, 0, stream>>>(w, Wb);
  sparsify_compress_kernel<<<dim3((M_TOTAL * (K_FEAT / 32)) / 256), 256, 0, stream>>>(x, Ap, meta);
  swmmac_gemm_kernel<<<dim3(N_FEAT / BN, M_TOTAL / BM), 256, 0, stream>>>(Ap, meta, Wb, bias, out);
}